// TransformerBitwise_62380105007446
// MI455X (gfx1250) — compile-verified
//
#include <hip/hip_runtime.h>
#include <stdint.h>

// out[i] = int64(a[i]) XOR int64(b[i]); a,b are exact integers < 2^20 in f32.
// Bandwidth-bound: 16 B/elem, 16 MB total -> ~0.7us at 23.3 TB/s (L2-resident
// on replay). Data movement uses the CDNA5 async global->LDS path (ASYNCcnt).

#define BLOCK_THREADS 256
#define TILE_ELEMS    1024   // 256 lanes * 4 floats (16 B per lane per operand)

// AMDGPU addrspace(3) pointers are 32-bit LDS byte offsets -- exactly the
// VDST operand format of global_load_async_to_lds_*.
typedef __attribute__((address_space(3))) uint8_t* lds_ptr_t;

__global__ __launch_bounds__(BLOCK_THREADS)
void TransformerBitwise_xor_kernel(const float* __restrict__ a,
                                   const float* __restrict__ b,
                                   long long* __restrict__ out,
                                   int n)
{
    // Layout: floats [0 .. TILE_ELEMS)        = a-tile (bytes 0    .. 4095)
    //         floats [TILE_ELEMS .. 2*TILE)   = b-tile (bytes 4096 .. 8191)
    __shared__ __align__(16) float smem[2 * TILE_ELEMS];

    const int tid  = threadIdx.x;
    const int base = blockIdx.x * TILE_ELEMS;
    const int idx  = base + tid * 4;           // this lane's first element
    // n % 4 == 0, idx % 4 == 0  =>  idx < n implies the whole float4 is valid.
    const bool active = (idx < n);

    if (active) {
        // Pass LDS destinations as addrspace(3) POINTERS so the allocation
        // escapes into the asm: the "memory" clobber then pins the later
        // ds_load reads (otherwise LLVM folds loads of never-stored LDS to
        // undef and deletes the whole compute phase -- seen in round 1).
        lds_ptr_t pa = (lds_ptr_t)&smem[tid * 4];
        lds_ptr_t pb = (lds_ptr_t)&smem[TILE_ELEMS + tid * 4];
        const uint64_t ga = (uint64_t)(uintptr_t)(a + idx);
        const uint64_t gb = (uint64_t)(uintptr_t)(b + idx);
        // CDNA5 async copy: 16 B per active lane, global -> LDS, ASYNCcnt-tracked.
        asm volatile(
            "global_load_async_to_lds_b128 %0, %2, off\n\t"
            "global_load_async_to_lds_b128 %1, %3, off"
            :
            : "v"(pa), "v"(pb), "v"(ga), "v"(gb)
            : "memory");
    }

    // Each wave drains its own async ops, then the split barrier publishes the
    // whole tile across the 8 waves of the workgroup.
    asm volatile("s_wait_asynccnt 0x0" ::: "memory");
    __syncthreads();

    if (active) {
        const float4 va = *reinterpret_cast<const float4*>(&smem[tid * 4]);
        const float4 vb = *reinterpret_cast<const float4*>(&smem[TILE_ELEMS + tid * 4]);

        // Exact: values are integers < 2^20, v_cvt_u32_f32 is lossless here.
        const uint32_t r0 = (uint32_t)va.x ^ (uint32_t)vb.x;
        const uint32_t r1 = (uint32_t)va.y ^ (uint32_t)vb.y;
        const uint32_t r2 = (uint32_t)va.z ^ (uint32_t)vb.z;
        const uint32_t r3 = (uint32_t)va.w ^ (uint32_t)vb.w;

        // 32 B of int64 output per lane as two b128 stores (16 B aligned).
        longlong2* o = reinterpret_cast<longlong2*>(out + idx);
        o[0] = make_longlong2((long long)r0, (long long)r1);
        o[1] = make_longlong2((long long)r2, (long long)r3);
    }
}

extern "C" void kernel_launch(void* const* d_in, const int* in_sizes, int n_in,
                              void* d_out, int out_size, void* d_ws, size_t ws_size,
                              hipStream_t stream)
{
    (void)n_in; (void)d_ws; (void)ws_size; (void)out_size;

    const float* a  = (const float*)d_in[0];
    const float* b  = (const float*)d_in[1];
    long long*  out = (long long*)d_out;       // reference output dtype: int64
    const int n = in_sizes[0];

    const int tiles = (n + TILE_ELEMS - 1) / TILE_ELEMS;   // 977 for n = 1e6
    TransformerBitwise_xor_kernel<<<dim3(tiles), dim3(BLOCK_THREADS), 0, stream>>>(
        a, b, out, n);
}